// NGCFLayer_4982162063610
// MI455X (gfx1250) — compile-verified
//
#include <hip/hip_runtime.h>
#include <hip/hip_bf16.h>

#define DIM 128
#define TILE_M 16

typedef __attribute__((ext_vector_type(2))) float v2f;
typedef __attribute__((ext_vector_type(8))) float v8f;

// ---------------------------------------------------------------------------
// Kernel 0: zero the neighbor accumulator (d_ws is poisoned, not zeroed)
// ---------------------------------------------------------------------------
__global__ void ngcf_zero(float* __restrict__ p, int n) {
    int i = blockIdx.x * blockDim.x + threadIdx.x;
    if (i < n) p[i] = 0.0f;
}

// ---------------------------------------------------------------------------
// Kernel 1: COO SpMM scatter.  One wave per edge; each of the 32 lanes owns a
// float4 chunk of the 128-dim row.  Gather from emb[col] (L2-resident: 51 MB
// of embeddings < 192 MB L2) and hardware-atomic-add into neighbor[row].
// ---------------------------------------------------------------------------
__global__ __launch_bounds__(256) void ngcf_scatter(
    const float* __restrict__ emb,
    const float* __restrict__ vals,
    const long long* __restrict__ rows,
    const long long* __restrict__ cols,
    float* __restrict__ nei,
    int n_edges)
{
    long long tid  = (long long)blockIdx.x * blockDim.x + threadIdx.x;
    int       e    = (int)(tid >> 5);
    int       lane = (int)(tid & 31);
    if (e >= n_edges) return;

    long long r = rows[e];
    long long c = cols[e];
    float     v = vals[e];

    const float4 m = *(const float4*)(emb + c * DIM + lane * 4);
    float* dst = nei + r * DIM + lane * 4;
    unsafeAtomicAdd(dst + 0, v * m.x);   // global_atomic_add_f32
    unsafeAtomicAdd(dst + 1, v * m.y);
    unsafeAtomicAdd(dst + 2, v * m.z);
    unsafeAtomicAdd(dst + 3, v * m.w);
}

// ---------------------------------------------------------------------------
// WMMA helper: one K=4 step of D = A*B + C using V_WMMA_F32_16X16X4_F32.
// ---------------------------------------------------------------------------
__device__ __forceinline__ v8f wmma_f32_k4(v2f a, v2f b, v8f c) {
    return __builtin_amdgcn_wmma_f32_16x16x4_f32(
        /*neg_a=*/false, a, /*neg_b=*/false, b,
        /*c_mod=*/(short)0, c, /*reuse_a=*/false, /*reuse_b=*/false);
}

// ---------------------------------------------------------------------------
// Kernel 2: fused dense stage.  Per block: 16 nodes x 128 dims.
// 8 waves, wave w owns output column tile [16w, 16w+16).
//   phase 1: neighbor_emb = nei @ W2^T + b2        (32 WMMA / wave)
//   phase 2: self = emb @ W1^T ; inter = prod @ W2^T  (64 WMMA / wave)
//   out = leaky_relu(self + b1 + neighbor_emb + inter + b2, 0.2)
// ---------------------------------------------------------------------------
__global__ __launch_bounds__(256) void ngcf_dense(
    const float* __restrict__ emb,
    const float* __restrict__ nei,
    const float* __restrict__ W1, const float* __restrict__ b1,
    const float* __restrict__ W2, const float* __restrict__ b2,
    float* __restrict__ out, int n_nodes)
{
    __shared__ float sE[TILE_M * DIM];   // embedding tile
    __shared__ float sN[TILE_M * DIM];   // neighbor tile, later prod tile
    __shared__ float sM[TILE_M * DIM];   // neighbor_emb (+b2) tile

    const int t     = threadIdx.x;
    const int node0 = blockIdx.x * TILE_M;

    // cooperative, coalesced tile load
    for (int i = t; i < TILE_M * DIM; i += 256) {
        int g = node0 + (i >> 7);
        float e = 0.0f, n = 0.0f;
        if (g < n_nodes) {
            long long off = (long long)g * DIM + (i & (DIM - 1));
            e = emb[off];
            n = nei[off];
        }
        sE[i] = e;
        sN[i] = n;
    }
    __syncthreads();

    const int wave = t >> 5;              // n-tile index 0..7
    const int lane = t & 31;
    const int half = lane >> 4;           // 0: K={k0,k0+1}, 1: K={k0+2,k0+3}
    const int l15  = lane & 15;           // A row (m) / B column (n)
    const int kk   = half * 2;
    const int ncol = wave * 16 + l15;     // global output column for B frags

    const float* w1row = W1 + (long long)ncol * DIM;  // B[k][n] = W[n][k]
    const float* w2row = W2 + (long long)ncol * DIM;
    const float  b1c   = b1[ncol];
    const float  b2c   = b2[ncol];

    // ---- phase 1: neighbor_emb tile ----
    v8f accN = {};
    #pragma unroll
    for (int k0 = 0; k0 < DIM; k0 += 4) {
        v2f a, b;
        a.x = sN[l15 * DIM + k0 + kk];
        a.y = sN[l15 * DIM + k0 + kk + 1];
        b.x = w2row[k0 + kk];
        b.y = w2row[k0 + kk + 1];
        accN = wmma_f32_k4(a, b, accN);
    }
    #pragma unroll
    for (int r = 0; r < 8; ++r) {
        int row = r + half * 8;           // C/D layout: VGPR r -> M=r (+8 hi half)
        sM[row * DIM + ncol] = accN[r] + b2c;
    }
    __syncthreads();

    // ---- elementwise interact input: prod = neighbor_emb * emb ----
    for (int i = t; i < TILE_M * DIM; i += 256) sN[i] = sM[i] * sE[i];
    __syncthreads();

    // ---- phase 2: self GEMM (W1) and interact GEMM (W2) ----
    v8f accS = {}, accI = {};
    #pragma unroll
    for (int k0 = 0; k0 < DIM; k0 += 4) {
        v2f aS, aI, bS, bI;
        aS.x = sE[l15 * DIM + k0 + kk];
        aS.y = sE[l15 * DIM + k0 + kk + 1];
        bS.x = w1row[k0 + kk];
        bS.y = w1row[k0 + kk + 1];
        accS = wmma_f32_k4(aS, bS, accS);

        aI.x = sN[l15 * DIM + k0 + kk];
        aI.y = sN[l15 * DIM + k0 + kk + 1];
        bI.x = w2row[k0 + kk];
        bI.y = w2row[k0 + kk + 1];
        accI = wmma_f32_k4(aI, bI, accI);
    }

    #pragma unroll
    for (int r = 0; r < 8; ++r) {
        int row = r + half * 8;
        int g   = node0 + row;
        if (g < n_nodes) {
            float v = accS[r] + b1c + accI[r] + b2c + sM[row * DIM + ncol];
            out[(long long)g * DIM + ncol] = v > 0.0f ? v : 0.2f * v;
        }
    }
}

// ---------------------------------------------------------------------------
// Launch
// inputs: 0 embeddings[N,128] f32, 1 adj_vals[E] f32, 2 W1[128,128] f32,
//         3 b1[128] f32, 4 W2[128,128] f32, 5 b2[128] f32,
//         6 adj_rows[E] i64, 7 adj_cols[E] i64
// ---------------------------------------------------------------------------
extern "C" void kernel_launch(void* const* d_in, const int* in_sizes, int n_in,
                              void* d_out, int out_size, void* d_ws, size_t ws_size,
                              hipStream_t stream) {
    const float*     emb  = (const float*)d_in[0];
    const float*     vals = (const float*)d_in[1];
    const float*     W1   = (const float*)d_in[2];
    const float*     b1   = (const float*)d_in[3];
    const float*     W2   = (const float*)d_in[4];
    const float*     b2   = (const float*)d_in[5];
    const long long* rows = (const long long*)d_in[6];
    const long long* cols = (const long long*)d_in[7];
    float*           out  = (float*)d_out;
    float*           nei  = (float*)d_ws;          // [N,128] f32 accumulator

    const int n_nodes = in_sizes[0] / DIM;
    const int n_edges = in_sizes[1];

    // 0) zero neighbor accumulator
    int ztot = n_nodes * DIM;
    ngcf_zero<<<(ztot + 255) / 256, 256, 0, stream>>>(nei, ztot);

    // 1) COO scatter: one wave (32 lanes x float4) per edge
    long long sthreads = (long long)n_edges * 32;
    int sblocks = (int)((sthreads + 255) / 256);
    ngcf_scatter<<<sblocks, 256, 0, stream>>>(emb, vals, rows, cols, nei, n_edges);

    // 2) fused WMMA dense stage: 16 nodes per block
    int dblocks = (n_nodes + TILE_M - 1) / TILE_M;
    ngcf_dense<<<dblocks, 256, 0, stream>>>(emb, nei, W1, b1, W2, b2, out, n_nodes);
}